// MemContrastiveLoss_32555852104025
// MI455X (gfx1250) — compile-verified
//
#include <hip/hip_runtime.h>
#include <hip/hip_bf16.h>
#include <math.h>

// OIM-style memory-contrastive loss, fused, split-M for occupancy.
// B=8, Fr=16, O=16, D=256, M=256.
// Grid: 128 (b,f) x 8 m-splits = 1024 blocks x 8 waves = 8192 waves in flight.
// Memory-bound (537 MB @ 23.3 TB/s ~ 23 us); f32 WMMA 16x16x4 for exact dots.

typedef __attribute__((ext_vector_type(2))) float v2f;
typedef __attribute__((ext_vector_type(8))) float v8f;

#define TEMP_INV    (1.0f / 0.07f)
#define O_DIM       16
#define D_DIM       256
#define M_DIM       256
#define M_SPLITS    8
#define M_PER_BLK   (M_DIM / M_SPLITS)      // 32
#define M_PER_WAVE  (M_PER_BLK / 8)         // 4  (waves/block = 8)
#define NBF         128                     // B*Fr
// workspace layout (floats): [0,2048) = sumexp[bf*16+i], [2048,4096) = possim
#define WS_SUM_OFF  0
#define WS_POS_OFF  (NBF * O_DIM)
#define WS_FLOATS   (2 * NBF * O_DIM + 1)

__global__ void oim_zero_ws(float* ws, float* out) {
    int idx = blockIdx.x * blockDim.x + threadIdx.x;
    if (idx < 2 * NBF * O_DIM) ws[idx] = 0.0f;
    if (idx == 0) out[0] = 0.0f;
}

__global__ void __launch_bounds__(256)
MemContrastiveLoss_oim_main(const float* __restrict__ preds,
                            const float* __restrict__ memry,
                            const int*   __restrict__ mem_table,
                            float* __restrict__ ws)
{
    __shared__ float lds_preds[O_DIM * D_DIM];   // 16 KB normalized pred tile
    __shared__ float lds_sum[O_DIM];             // block-partial sum of exp
    __shared__ int   lds_pc[O_DIM];              // per-i pos_count

    const int bf     = blockIdx.x;               // 0..127
    const int sp     = blockIdx.y;               // m-split 0..7
    const int tid    = threadIdx.x;
    const int wave   = tid >> 5;
    const int lane   = tid & 31;
    const int hv     = lane >> 4;                // lane half (0/1)
    const int lane16 = lane & 15;

    const float* pred_bf = preds + (size_t)bf * (O_DIM * D_DIM);
    const float* mem_bf  = memry + (size_t)bf * ((size_t)M_DIM * O_DIM * D_DIM);

    // ---- Phase 1: L2-normalize pred rows into LDS (one row per thread 0..15) ----
    if (tid < O_DIM) {
        const float* row = pred_bf + tid * D_DIM;
        float n2 = 0.0f;
        for (int d = 0; d < D_DIM; ++d) { float v = row[d]; n2 += v * v; }
        const float inv = 1.0f / fmaxf(sqrtf(n2), 1e-12f);
        float* dst = lds_preds + tid * D_DIM;
        for (int d = 0; d < D_DIM; ++d) dst[d] = row[d] * inv;
        lds_sum[tid] = 0.0f;
        lds_pc[tid]  = mem_table[bf * O_DIM + tid];
    }
    __syncthreads();

    int pcv[8];                                  // pos_count for i = r + 8*hv
    #pragma unroll
    for (int r = 0; r < 8; ++r) pcv[r] = lds_pc[r + 8 * hv];

    float acc[8];
    #pragma unroll
    for (int r = 0; r < 8; ++r) acc[r] = 0.0f;

    // A-matrix (16x4 f32): lane holds M = lane16, K = k0 + 2*hv + {0,1}
    const v2f* aBase = reinterpret_cast<const v2f*>(lds_preds);
    const int  aRow  = (lane16 * D_DIM + 2 * hv) >> 1;

    const int mbase = sp * M_PER_BLK + wave * M_PER_WAVE;

    // ---- Phase 2: two m-tiles at a time -> two independent WMMA/load chains ----
    for (int pair = 0; pair < M_PER_WAVE / 2; ++pair) {
        const int m0 = mbase + 2 * pair;
        const int m1 = m0 + 1;
        const float* brow0 = mem_bf + ((size_t)m0 * O_DIM + lane16) * D_DIM + 2 * hv;
        const float* brow1 = brow0 + (size_t)O_DIM * D_DIM;

        if (pair + 1 < M_PER_WAVE / 2)
            __builtin_prefetch(brow1 + (size_t)O_DIM * D_DIM, 0, 1);  // next pair

        v8f   c0 = {}, c1 = {};
        float nsq0 = 0.0f, nsq1 = 0.0f;
        #pragma unroll 8
        for (int k0 = 0; k0 < D_DIM; k0 += 4) {
            v2f a  = aBase[aRow + (k0 >> 1)];                     // shared by both tiles
            v2f b0 = *reinterpret_cast<const v2f*>(brow0 + k0);
            v2f b1 = *reinterpret_cast<const v2f*>(brow1 + k0);
            nsq0 += b0.x * b0.x + b0.y * b0.y;
            nsq1 += b1.x * b1.x + b1.y * b1.y;
            c0 = __builtin_amdgcn_wmma_f32_16x16x4_f32(false, a, false, b0,
                                                       (short)0, c0, false, false);
            c1 = __builtin_amdgcn_wmma_f32_16x16x4_f32(false, a, false, b1,
                                                       (short)0, c1, false, false);
        }

        // full ||mem[m,j,:]||^2: lane halves hold disjoint K subsets of row j
        nsq0 += __shfl_xor(nsq0, 16);
        nsq1 += __shfl_xor(nsq1, 16);
        const float scale0 = TEMP_INV / fmaxf(sqrtf(nsq0), 1e-12f);
        const float scale1 = TEMP_INV / fmaxf(sqrtf(nsq1), 1e-12f);

        // C/D layout: element (i = r + 8*hv, j = lane16) in c[r]
        #pragma unroll
        for (int r = 0; r < 8; ++r) {
            const int   i  = r + 8 * hv;
            const int   j  = lane16;
            const float s0 = c0[r] * scale0;
            const float s1 = c1[r] * scale1;
            const bool  isPos  = (m0 == 0) & (j == i);   // only sp=0,wave=0,pair=0
            const bool  valid0 = ((m0 < pcv[r]) & (j != i)) | isPos;
            const bool  valid1 = (m1 < pcv[r]) & (j != i);
            if (isPos) ws[WS_POS_OFF + bf * O_DIM + i] = s0;   // unique writer
            acc[r] += (valid0 ? __expf(s0) : 0.0f) + (valid1 ? __expf(s1) : 0.0f);
        }
    }

    // ---- Phase 3: reduce over the 16 j-lanes of each half, then to workspace ----
    #pragma unroll
    for (int r = 0; r < 8; ++r) {
        float v = acc[r];
        v += __shfl_xor(v, 1);
        v += __shfl_xor(v, 2);
        v += __shfl_xor(v, 4);
        v += __shfl_xor(v, 8);
        if (lane16 == 0) atomicAdd(&lds_sum[r + 8 * hv], v);
    }
    __syncthreads();

    if (tid < O_DIM)
        atomicAdd(&ws[WS_SUM_OFF + bf * O_DIM + tid], lds_sum[tid]);
}

__global__ void oim_loss_final(const float* __restrict__ ws, float* __restrict__ out) {
    int idx = blockIdx.x * blockDim.x + threadIdx.x;   // (bf,i) flattened, 0..2047
    if (idx < NBF * O_DIM) {
        const float lp = ws[WS_POS_OFF + idx] - __logf(ws[WS_SUM_OFF + idx]);
        const float p  = __expf(lp);
        const float q  = 1.0f - p + 1e-12f;
        const float li = -(q * q) * __logf(p + 1e-12f);
        atomicAdd(out, li * (1.0f / (float)NBF));      // / (B*Fr)
    }
}

extern "C" void kernel_launch(void* const* d_in, const int* in_sizes, int n_in,
                              void* d_out, int out_size, void* d_ws, size_t ws_size,
                              hipStream_t stream) {
    const float* preds = (const float*)d_in[0];   // (8,16,16,256) f32
    const float* mem   = (const float*)d_in[1];   // (8,16,256,16,256) f32
    const int*   mt    = (const int*)d_in[2];     // (8,16,16) i32
    float*       out   = (float*)d_out;           // scalar f32
    float*       ws    = (float*)d_ws;            // >= WS_FLOATS * 4 bytes

    oim_zero_ws<<<(2 * NBF * O_DIM + 255) / 256, 256, 0, stream>>>(ws, out);

    dim3 grid(NBF, M_SPLITS);
    MemContrastiveLoss_oim_main<<<grid, 256, 0, stream>>>(preds, mem, mt, ws);

    oim_loss_final<<<(NBF * O_DIM + 255) / 256, 256, 0, stream>>>(ws, out);
}